// PointLSTM_71751723647267
// MI455X (gfx1250) — compile-verified
//
#include <hip/hip_runtime.h>
#include <math.h>

typedef float v2f __attribute__((ext_vector_type(2)));
typedef float v8f __attribute__((ext_vector_type(8)));

#define BATCH 8
#define NPTS  512
#define SEQL  10
#define TPB   256

// ---------------------------------------------------------------------------
// Farthest point sampling: one block per batch, serial npoint loop with a
// block-wide argmax reduction (tie -> smallest index, matching jnp.argmax).
// ---------------------------------------------------------------------------
__global__ void fps_kernel(const float* __restrict__ xyz, int bstride, int Np,
                           int npoint, int* __restrict__ outIdx) {
  __shared__ float dist[512];
  __shared__ float sd[TPB];
  __shared__ int   si[TPB];
  __shared__ int   s_last;
  int b = blockIdx.x;
  const float* p = xyz + (size_t)b * bstride;
  int t = threadIdx.x;
  for (int i = t; i < Np; i += TPB) dist[i] = 1e10f;
  if (t == 0) { outIdx[b * npoint] = 0; s_last = 0; }
  __syncthreads();
  for (int it = 1; it < npoint; ++it) {
    int last = s_last;
    float px = p[last*3+0], py = p[last*3+1], pz = p[last*3+2];
    float best = -1.0f; int bi = 0x7fffffff;
    for (int i = t; i < Np; i += TPB) {
      float dx = p[i*3+0]-px, dy = p[i*3+1]-py, dz = p[i*3+2]-pz;
      float d = dx*dx + dy*dy + dz*dz;
      float nd = fminf(dist[i], d);
      dist[i] = nd;
      if (nd > best) { best = nd; bi = i; }   // first max within strided set
    }
    sd[t] = best; si[t] = bi;
    __syncthreads();
    for (int s = TPB/2; s > 0; s >>= 1) {
      if (t < s) {
        if (sd[t+s] > sd[t] || (sd[t+s] == sd[t] && si[t+s] < si[t])) {
          sd[t] = sd[t+s]; si[t] = si[t+s];
        }
      }
      __syncthreads();
    }
    if (t == 0) { s_last = si[0]; outIdx[b*npoint + it] = si[0]; }
    __syncthreads();
  }
}

__global__ void gather_xyz(const float* __restrict__ src, int bstride,
                           const int* __restrict__ idx, int M,
                           float* __restrict__ dst) {
  int g = blockIdx.x * blockDim.x + threadIdx.x;
  if (g >= BATCH * M * 3) return;
  int d = g % 3; int bm = g / 3; int m = bm % M; int b = bm / M;
  dst[g] = src[(size_t)b * bstride + (size_t)idx[b*M + m] * 3 + d];
}

__global__ void slice_copy(const float* __restrict__ xyzs, int ti,
                           float* __restrict__ frame) {
  int g = blockIdx.x * blockDim.x + threadIdx.x;
  if (g >= BATCH * NPTS * 3) return;
  int b = g / (NPTS*3); int r = g % (NPTS*3);
  frame[g] = xyzs[((size_t)b * SEQL + ti) * NPTS * 3 + r];
}

// ---------------------------------------------------------------------------
// Ball query: first `nsample` indices (ascending) with d2 < r2, pad with the
// first hit (or 0 if none) — equivalent to the argsort-by-key reference.
// ---------------------------------------------------------------------------
__global__ void ball_query_kernel(const float* __restrict__ xyz,
                                  const float* __restrict__ nxyz,
                                  int Nsrc, int M, int nsample, float r2,
                                  int* __restrict__ idx) {
  int g = blockIdx.x * blockDim.x + threadIdx.x;
  if (g >= BATCH * M) return;
  int b = g / M;
  const float* src = xyz + (size_t)b * Nsrc * 3;
  float qx = nxyz[g*3+0], qy = nxyz[g*3+1], qz = nxyz[g*3+2];
  int* o = idx + (size_t)g * nsample;
  int cnt = 0, first = 0;
  for (int j = 0; j < Nsrc && cnt < nsample; ++j) {
    float dx = src[j*3+0]-qx, dy = src[j*3+1]-qy, dz = src[j*3+2]-qz;
    float d2 = dx*dx + dy*dy + dz*dz;
    if (d2 < r2) { if (cnt == 0) first = j; o[cnt++] = j; }
  }
  for (; cnt < nsample; ++cnt) o[cnt] = first;
}

// ---------------------------------------------------------------------------
// Pack concat([H2g, X1, disp]) -> row-major (rows, ginpad), zero K-padding.
// ---------------------------------------------------------------------------
__global__ void build_hcat(const float* __restrict__ H2,
                           const float* __restrict__ X1,
                           const float* __restrict__ P2,
                           const float* __restrict__ P1,
                           const int* __restrict__ idx,
                           int M, int k, int outc, int inc, int ginpad,
                           float* __restrict__ hcat) {
  int g = blockIdx.x * blockDim.x + threadIdx.x;
  int rows = BATCH * M * k;
  if (g >= rows * ginpad) return;
  int c = g % ginpad; int row = g / ginpad;
  int bm = row / k; int m = bm % M; int b = bm / M;
  int j = idx[row];
  float v = 0.0f;
  if (c < outc) {
    v = H2[((size_t)b * M + j) * outc + c];
  } else if (c < outc + inc) {
    v = X1[((size_t)b * M + m) * inc + (c - outc)];
  } else if (c < outc + inc + 3) {
    int d = c - outc - inc;
    v = P2[((size_t)b * M + j) * 3 + d] - P1[((size_t)b * M + m) * 3 + d];
  }
  hcat[g] = v;
}

// ---------------------------------------------------------------------------
// Pre-pack weights: zero-pad K to Kpad and interleave row pairs as float2 so
// the GEMM's B-fragment is a single aligned b64 load per lane per K-quad.
//   Wp[r*N + c] = { W[2r][c], W[2r+1][c] }   (0 beyond Kreal)
// ---------------------------------------------------------------------------
__global__ void pack_w(const float* __restrict__ W, int Kreal, int Kpad, int N,
                       v2f* __restrict__ Wp) {
  int g = blockIdx.x * blockDim.x + threadIdx.x;
  int total = (Kpad >> 1) * N;
  if (g >= total) return;
  int c = g % N; int r = g / N;
  int k0 = 2*r, k1 = 2*r + 1;
  v2f v;
  v.x = (k0 < Kreal) ? W[(size_t)k0 * N + c] : 0.0f;
  v.y = (k1 < Kreal) ? W[(size_t)k1 * N + c] : 0.0f;
  Wp[g] = v;
}

// ---------------------------------------------------------------------------
// FP32 WMMA GEMM: C[rows,N] = A[rows,Kpad] @ Wp (+bias)(+relu).
// One wave per 16x16 tile, K stepped by 4 via V_WMMA_F32_16X16X4_F32.
// rows % 16 == 0, N % 16 == 0, Kpad % 4 == 0; A pad columns and Wp pad rows
// are zero, so no bounds checks / EXEC manipulation in the inner loop.
// ---------------------------------------------------------------------------
__global__ void wmma_gemm(const float* __restrict__ A,
                          const v2f* __restrict__ Wp,
                          float* __restrict__ C,
                          int N, int Kpad,
                          const float* __restrict__ bias, int act) {
  int tn = blockIdx.x * 16;
  int tm = blockIdx.y * 16;
  int lane = threadIdx.x;
  int half = lane >> 4;      // 0: K+0/K+1, 1: K+2/K+3
  int l16  = lane & 15;
  v8f acc = {};
  const float* arow = A + (size_t)(tm + l16) * Kpad + half * 2;
  const v2f*   bcol = Wp + (size_t)half * N + (tn + l16);
  for (int k = 0; k < Kpad; k += 4) {
    v2f a = *(const v2f*)(arow + k);
    v2f b = bcol[(size_t)(k >> 1) * N];
    acc = __builtin_amdgcn_wmma_f32_16x16x4_f32(false, a, false, b,
                                                (short)0, acc, false, false);
  }
  for (int r = 0; r < 8; ++r) {
    int row = tm + r + half * 8;
    int col = tn + l16;
    float v = acc[r];
    if (bias) v += bias[col];
    if (act == 1) v = fmaxf(v, 0.0f);
    C[(size_t)row * N + col] = v;
  }
}

// max over k neighbors, + bias, then activation (0 none, 1 sigmoid, 2 tanh)
__global__ void gate_reduce(const float* __restrict__ Y,
                            const float* __restrict__ bias,
                            int M, int k, int outc, int act,
                            float* __restrict__ G) {
  int g = blockIdx.x * blockDim.x + threadIdx.x;
  if (g >= BATCH * M * outc) return;
  int c = g % outc; int bm = g / outc;
  float best = -1e30f;
  for (int kk = 0; kk < k; ++kk)
    best = fmaxf(best, Y[((size_t)bm * k + kk) * outc + c]);
  best += bias[c];
  if (act == 1)      best = 1.0f / (1.0f + expf(-best));
  else if (act == 2) best = tanhf(best);
  G[g] = best;
}

__global__ void lstm_combine(const float* __restrict__ I,
                             const float* __restrict__ F,
                             const float* __restrict__ O,
                             const float* __restrict__ Cn,
                             const float* __restrict__ Co,
                             float* __restrict__ H, float* __restrict__ C,
                             int n) {
  int g = blockIdx.x * blockDim.x + threadIdx.x;
  if (g >= n) return;
  float c = F[g] * Co[g] + I[g] * Cn[g];
  C[g] = c;
  H[g] = O[g] * tanhf(c);
}

__global__ void feat_pool(const float* __restrict__ H,
                          const int* __restrict__ idx,
                          int M, int ns, int Nsrc, int C,
                          float* __restrict__ outF) {
  int g = blockIdx.x * blockDim.x + threadIdx.x;
  if (g >= BATCH * M * C) return;
  int c = g % C; int bm = g / C; int m = bm % M; int b = bm / M;
  const int* ir = idx + (size_t)(b * M + m) * ns;
  float best = -1e30f;
  for (int kk = 0; kk < ns; ++kk)
    best = fmaxf(best, H[((size_t)b * Nsrc + ir[kk]) * C + c]);
  outF[g] = best;
}

// three nearest neighbors + normalized inverse-distance weights
__global__ void three_nn(const float* __restrict__ unk,
                         const float* __restrict__ kn,
                         int Nn, int Nk, int* __restrict__ idx3,
                         float* __restrict__ w3) {
  int g = blockIdx.x * blockDim.x + threadIdx.x;
  if (g >= BATCH * Nn) return;
  int b = g / Nn;
  float qx = unk[g*3+0], qy = unk[g*3+1], qz = unk[g*3+2];
  const float* K = kn + (size_t)b * Nk * 3;
  float d0 = 1e30f, d1 = 1e30f, d2 = 1e30f;
  int i0 = 0, i1 = 0, i2 = 0;
  for (int m = 0; m < Nk; ++m) {
    float dx = K[m*3+0]-qx, dy = K[m*3+1]-qy, dz = K[m*3+2]-qz;
    float d = dx*dx + dy*dy + dz*dz;
    if (d < d0)      { d2=d1; i2=i1; d1=d0; i1=i0; d0=d; i0=m; }
    else if (d < d1) { d2=d1; i2=i1; d1=d;  i1=m; }
    else if (d < d2) { d2=d;  i2=m; }
  }
  float w0 = 1.0f/(d0+1e-8f), w1 = 1.0f/(d1+1e-8f), w2 = 1.0f/(d2+1e-8f);
  float s = w0 + w1 + w2;
  idx3[g*3+0] = i0; idx3[g*3+1] = i1; idx3[g*3+2] = i2;
  w3[g*3+0] = w0/s; w3[g*3+1] = w1/s; w3[g*3+2] = w2/s;
}

__global__ void interp_apply(const float* __restrict__ F,
                             const int* __restrict__ idx3,
                             const float* __restrict__ w3,
                             int Nn, int Nk, int Cf,
                             float* __restrict__ dst, int ldd, int off) {
  int g = blockIdx.x * blockDim.x + threadIdx.x;
  if (g >= BATCH * Nn * Cf) return;
  int c = g % Cf; int bn = g / Cf; int b = bn / Nn;
  const int*   id = idx3 + (size_t)bn * 3;
  const float* w  = w3   + (size_t)bn * 3;
  const float* Fb = F + (size_t)b * Nk * Cf;
  float v = w[0]*Fb[(size_t)id[0]*Cf + c]
          + w[1]*Fb[(size_t)id[1]*Cf + c]
          + w[2]*Fb[(size_t)id[2]*Cf + c];
  dst[(size_t)bn * ldd + off + c] = v;
}

__global__ void copy_cols(const float* __restrict__ src, int Nn, int Cs,
                          float* __restrict__ dst, int ldd, int off) {
  int g = blockIdx.x * blockDim.x + threadIdx.x;
  if (g >= BATCH * Nn * Cs) return;
  int c = g % Cs; int bn = g / Cs;
  dst[(size_t)bn * ldd + off + c] = src[g];
}

// motion = h @ w2 + b2 ; frame += motion ; write frame into output slice
__global__ void mlp2_update(const float* __restrict__ h,
                            const float* __restrict__ w2,
                            const float* __restrict__ b2,
                            float* __restrict__ frame,
                            float* __restrict__ out, int ti) {
  int g = blockIdx.x * blockDim.x + threadIdx.x;
  if (g >= BATCH * NPTS) return;
  int b = g / NPTS; int n = g % NPTS;
  const float* hr = h + (size_t)g * 64;
  float a0 = b2[0], a1 = b2[1], a2 = b2[2];
  for (int i = 0; i < 64; ++i) {
    float hv = hr[i];
    a0 = fmaf(hv, w2[i*3+0], a0);
    a1 = fmaf(hv, w2[i*3+1], a1);
    a2 = fmaf(hv, w2[i*3+2], a2);
  }
  float* fr = frame + (size_t)g * 3;
  float f0 = fr[0] + a0, f1 = fr[1] + a1, f2 = fr[2] + a2;
  fr[0] = f0; fr[1] = f1; fr[2] = f2;
  float* o = out + (((size_t)b * (SEQL/2) + ti) * NPTS + n) * 3;
  o[0] = f0; o[1] = f1; o[2] = f2;
}

// ---------------------------------------------------------------------------
#define GRID1(n) dim3(((n) + TPB - 1) / TPB), dim3(TPB), 0, stream

extern "C" void kernel_launch(void* const* d_in, const int* in_sizes, int n_in,
                              void* d_out, int out_size, void* d_ws, size_t ws_size,
                              hipStream_t stream) {
  (void)in_sizes; (void)n_in; (void)out_size; (void)ws_size;
  const float* xyzs   = (const float*)d_in[0];
  const float* mlp_w1 = (const float*)d_in[61];
  const float* mlp_b1 = (const float*)d_in[62];
  const float* mlp_w2 = (const float*)d_in[63];
  const float* mlp_b2 = (const float*)d_in[64];
  float* out = (float*)d_out;

  // --- workspace bump allocator (deterministic layout) ---
  size_t off = 0;
  char* base = (char*)d_ws;
  auto allocf = [&](size_t n) -> float* {
    size_t a = (off + 255) & ~(size_t)255; off = a + n * sizeof(float);
    return (float*)(base + a);
  };
  auto alloci = [&](size_t n) -> int* {
    size_t a = (off + 255) & ~(size_t)255; off = a + n * sizeof(int);
    return (int*)(base + a);
  };

  const int   Mlv[3]      = {256, 128, 64};
  const int   klv[3]      = {12, 8, 4};
  const int   outlv[3]    = {64, 128, 256};
  const int   inlv[3]     = {0, 64, 128};
  const int   ginpadlv[3] = {68, 196, 388};   // (out+in+3) rounded up to 4
  const int   oinpadlv[3] = {68, 132, 260};   // (out+3) rounded up to 4
  const float rlv[3]      = {4.000001f, 8.000001f, 12.000001f};
  const int   srcNlv[3]   = {512, 256, 128};
  const float poolr[3]    = {0.0f, 2.000001f, 4.000001f};

  float* frame = allocf((size_t)BATCH * NPTS * 3);
  float* xyzA[3]; float* xyzB[3];
  for (int l = 0; l < 3; ++l) {
    xyzA[l] = allocf((size_t)BATCH * Mlv[l] * 3);
    xyzB[l] = allocf((size_t)BATCH * Mlv[l] * 3);
  }
  float* Hst[3][2]; float* Cst[3][2];
  for (int l = 0; l < 3; ++l)
    for (int s = 0; s < 2; ++s) {
      Hst[l][s] = allocf((size_t)BATCH * Mlv[l] * outlv[l]);
      Cst[l][s] = allocf((size_t)BATCH * Mlv[l] * outlv[l]);
    }
  int*   fpsIdx = alloci((size_t)BATCH * 256);
  int*   idxL   = alloci((size_t)BATCH * 256 * 12);
  int*   idxP   = alloci((size_t)BATCH * 128 * 4);
  float* hcat   = allocf((size_t)24576 * 68);           // max packed activ.
  float* Y      = allocf((size_t)24576 * 64);           // max GEMM output
  float* G[5];
  for (int i = 0; i < 5; ++i) G[i] = allocf((size_t)BATCH * 256 * 64);
  float* feat2 = allocf((size_t)BATCH * 128 * 64);
  float* feat3 = allocf((size_t)BATCH * 64 * 128);
  int*   idx3  = alloci((size_t)BATCH * 512 * 3);
  float* w3    = allocf((size_t)BATCH * 512 * 3);
  float* l3    = allocf((size_t)BATCH * 128 * 384);
  float* l2    = allocf((size_t)BATCH * 256 * 448);
  float* l1    = allocf((size_t)BATCH * 512 * 448);

  // --- pre-pack all GEMM weights into padded float2-interleaved layout ---
  v2f* Wp[6][5];      // [cell: en1,en2,en3,de1,de2,de3][gate: i,f,o,c,old]
  for (int ci = 0; ci < 6; ++ci) {
    int l  = ci % 3;
    int oc = outlv[l];
    for (int g = 0; g < 4; ++g)
      Wp[ci][g] = (v2f*)allocf((size_t)ginpadlv[l] * oc);
    Wp[ci][4] = (v2f*)allocf((size_t)oinpadlv[l] * oc);
  }
  v2f* mlpw1p = (v2f*)allocf((size_t)448 * 64);
  for (int ci = 0; ci < 6; ++ci) {
    int l   = ci % 3;
    int oc  = outlv[l];
    int gin = oc + inlv[l] + 3, oin = oc + 3;
    int cb  = 1 + ci * 10;
    for (int g = 0; g < 4; ++g) {
      int n = (ginpadlv[l] >> 1) * oc;
      pack_w<<<GRID1(n)>>>((const float*)d_in[cb + g*2], gin, ginpadlv[l], oc,
                           Wp[ci][g]);
    }
    { int n = (oinpadlv[l] >> 1) * oc;
      pack_w<<<GRID1(n)>>>((const float*)d_in[cb + 8], oin, oinpadlv[l], oc,
                           Wp[ci][4]); }
  }
  { int n = (448 >> 1) * 64;
    pack_w<<<GRID1(n)>>>(mlp_w1, 448, 448, 64, mlpw1p); }

  // state pointers (double buffered, swapped host-side per LSTM call)
  float* pP[3]; float* pX[3]; float* pH[3]; float* pHn[3]; float* pC[3]; float* pCn[3];
  for (int l = 0; l < 3; ++l) {
    pX[l] = xyzA[l]; pP[l] = xyzB[l];
    pH[l] = Hst[l][0]; pHn[l] = Hst[l][1];
    pC[l] = Cst[l][0]; pCn[l] = Cst[l][1];
  }

  for (int t = 0; t < SEQL; ++t) {
    bool enc = (t < SEQL/2);
    const float* fsrc; int fbstride;
    if (enc) {
      fsrc = xyzs + (size_t)t * NPTS * 3;
      fbstride = SEQL * NPTS * 3;
    } else {
      if (t == SEQL/2) {
        int n = BATCH * NPTS * 3;
        slice_copy<<<GRID1(n)>>>(xyzs, SEQL/2 - 1, frame);
      }
      fsrc = frame; fbstride = NPTS * 3;
    }

    for (int l = 0; l < 3; ++l) {
      int M = Mlv[l], k = klv[l], oc = outlv[l], ic = inlv[l];
      int gp = ginpadlv[l], op = oinpadlv[l];
      int ci = (enc ? 0 : 3) + l;
      int cb = 1 + ci * 10;
      const float* sx = (l == 0) ? fsrc : pP[l-1];
      int sbs = (l == 0) ? fbstride : Mlv[l-1] * 3;
      int sN = srcNlv[l];

      fps_kernel<<<dim3(BATCH), dim3(TPB), 0, stream>>>(sx, sbs, sN, M, fpsIdx);
      { int n = BATCH*M*3; gather_xyz<<<GRID1(n)>>>(sx, sbs, fpsIdx, M, pX[l]); }

      const float* featPtr = nullptr;
      if (l >= 1) {  // pooled feature from previous level's NEW hidden state
        int Mp = Mlv[l-1], Cp = outlv[l-1];
        float r2 = poolr[l] * poolr[l];
        { int n = BATCH*M;
          ball_query_kernel<<<GRID1(n)>>>(pP[l-1], pX[l], Mp, M, 4, r2, idxP); }
        float* fb = (l == 1) ? feat2 : feat3;
        { int n = BATCH*M*Cp;
          feat_pool<<<GRID1(n)>>>(pH[l-1], idxP, M, 4, Mp, Cp, fb); }
        featPtr = fb;
      }

      if (t == 0) {  // st == None: P2 = P1, H2 = C2 = 0
        hipMemcpyAsync(pP[l], pX[l], (size_t)BATCH*M*3*sizeof(float),
                       hipMemcpyDeviceToDevice, stream);
        hipMemsetAsync(pH[l], 0, (size_t)BATCH*M*oc*sizeof(float), stream);
        hipMemsetAsync(pC[l], 0, (size_t)BATCH*M*oc*sizeof(float), stream);
      }

      float r2 = rlv[l] * rlv[l];
      { int n = BATCH*M;
        ball_query_kernel<<<GRID1(n)>>>(pP[l], pX[l], M, M, k, r2, idxL); }

      int rows = BATCH * M * k;
      // gates i, f, o, c share hcat = [H2g | X1 | disp]
      { int n = rows * gp;
        build_hcat<<<GRID1(n)>>>(pH[l], featPtr, pP[l], pX[l], idxL,
                                 M, k, oc, ic, gp, hcat); }
      for (int g = 0; g < 4; ++g) {
        const float* bb = (const float*)d_in[cb + g*2 + 1];
        dim3 gg(oc/16, rows/16);
        wmma_gemm<<<gg, dim3(32), 0, stream>>>(hcat, Wp[ci][g], Y, oc, gp,
                                               nullptr, 0);
        int act = (g == 3) ? 2 : 1;  // c -> tanh, i/f/o -> sigmoid
        int n = BATCH*M*oc;
        gate_reduce<<<GRID1(n)>>>(Y, bb, M, k, oc, act, G[g]);
      }
      // old gate: hcat_old = [C2g | disp]
      { int n = rows * op;
        build_hcat<<<GRID1(n)>>>(pC[l], nullptr, pP[l], pX[l], idxL,
                                 M, k, oc, 0, op, hcat); }
      {
        const float* bb = (const float*)d_in[cb + 9];
        dim3 gg(oc/16, rows/16);
        wmma_gemm<<<gg, dim3(32), 0, stream>>>(hcat, Wp[ci][4], Y, oc, op,
                                               nullptr, 0);
        int n = BATCH*M*oc;
        gate_reduce<<<GRID1(n)>>>(Y, bb, M, k, oc, 0, G[4]);
      }
      { int n = BATCH*M*oc;
        lstm_combine<<<GRID1(n)>>>(G[0], G[1], G[2], G[3], G[4],
                                   pHn[l], pCn[l], n); }
      // swap: current xyz/H/C become the "previous state" for next timestep
      { float* tmp;
        tmp = pH[l]; pH[l] = pHn[l]; pHn[l] = tmp;
        tmp = pC[l]; pC[l] = pCn[l]; pCn[l] = tmp;
        tmp = pP[l]; pP[l] = pX[l];  pX[l]  = tmp; }
    }

    if (!enc) {  // decoder head: FP + MLP
      // l3 = concat(interp(x2, x3, h3), h2)
      { int n = BATCH*128; three_nn<<<GRID1(n)>>>(pP[1], pP[2], 128, 64, idx3, w3); }
      { int n = BATCH*128*256;
        interp_apply<<<GRID1(n)>>>(pH[2], idx3, w3, 128, 64, 256, l3, 384, 0); }
      { int n = BATCH*128*128; copy_cols<<<GRID1(n)>>>(pH[1], 128, 128, l3, 384, 256); }
      // l2 = concat(interp(x1, x2, l3), h1)
      { int n = BATCH*256; three_nn<<<GRID1(n)>>>(pP[0], pP[1], 256, 128, idx3, w3); }
      { int n = BATCH*256*384;
        interp_apply<<<GRID1(n)>>>(l3, idx3, w3, 256, 128, 384, l2, 448, 0); }
      { int n = BATCH*256*64; copy_cols<<<GRID1(n)>>>(pH[0], 256, 64, l2, 448, 384); }
      // l1 = interp(frame, x1, l2)
      { int n = BATCH*512; three_nn<<<GRID1(n)>>>(frame, pP[0], 512, 256, idx3, w3); }
      { int n = BATCH*512*448;
        interp_apply<<<GRID1(n)>>>(l2, idx3, w3, 512, 256, 448, l1, 448, 0); }
      // h = relu(l1 @ mlp_w1 + b1) via WMMA, then motion + frame update
      { dim3 gg(64/16, (BATCH*512)/16);
        wmma_gemm<<<gg, dim3(32), 0, stream>>>(l1, mlpw1p, Y, 64, 448,
                                               mlp_b1, 1); }
      { int n = BATCH*NPTS;
        mlp2_update<<<GRID1(n)>>>(Y, mlp_w2, mlp_b2, frame, out, t - SEQL/2); }
    }
  }
}